// SemiConvModel_42726334661121
// MI455X (gfx1250) — compile-verified
//
#include <hip/hip_runtime.h>

#define B_    16
#define CIN   3
#define H_    512
#define W_    512
#define COUT  16
#define K_    15
#define HALF  7

#define SROW  528          // LDS row stride in floats (multiple of 4 -> 16B)
#define BLOCK 128          // 4 wave32 waves; 128*4 = 512 cols per row

typedef float v4f __attribute__((ext_vector_type(4)));
typedef int   v4i __attribute__((ext_vector_type(4)));

#if __has_builtin(__builtin_amdgcn_global_load_async_to_lds_b128)
#define USE_ASYNC_LDS 1
typedef __attribute__((address_space(1))) v4i* gas4_t;   // global int4*
typedef __attribute__((address_space(3))) v4i* las4_t;   // LDS int4*
#else
#define USE_ASYNC_LDS 0
#endif

__global__ __launch_bounds__(BLOCK) void semiconv_maxplus_kernel(
    const float* __restrict__ x,      // [B, CIN, H, W]
    const float* __restrict__ scale_p,
    float* __restrict__ out)          // [B, COUT, H, W]
{
    __shared__ float s[CIN * SROW];

    const int t  = threadIdx.x;
    const int bh = blockIdx.x;            // 0 .. B*H-1
    const int h  = bh & (H_ - 1);
    const int b  = bh >> 9;               // / 512

    // ---------------- halo: -inf on both sides of each channel row ------
    // per channel: LDS idx 0..7 and 520..527 get -inf (interior sits at 8..519)
    if (t < CIN * 16) {
        const int c = t >> 4;
        const int p = t & 15;
        const int idx = (p < 8) ? p : (512 + p);  // 0..7 or 520..527
        s[c * SROW + idx] = -__builtin_inff();
    }

    // ---------------- stage interior rows into LDS ----------------------
    const float* rowbase = x + ((size_t)(b * CIN) * H_ + h) * W_;
#if USE_ASYNC_LDS
    #pragma unroll
    for (int c = 0; c < CIN; ++c) {
        const float* g = rowbase + (size_t)c * H_ * W_ + 4 * t;
        float*       l = &s[c * SROW + 8 + 4 * t];
        __builtin_amdgcn_global_load_async_to_lds_b128(
            (gas4_t)g, (las4_t)l, /*offset=*/0, /*cpol=*/0);
    }
    // make this wave's async LDS writes complete before the barrier
#if __has_builtin(__builtin_amdgcn_s_wait_asynccnt)
    __builtin_amdgcn_s_wait_asynccnt(0);
#else
    asm volatile("s_wait_asynccnt 0" ::: "memory");
#endif
#else
    #pragma unroll
    for (int c = 0; c < CIN; ++c) {
        const v4f v = *(const v4f*)(rowbase + (size_t)c * H_ * W_ + 4 * t);
        *(v4f*)&s[c * SROW + 8 + 4 * t] = v;
    }
#endif
    __syncthreads();

    // ---------------- parabolic weights ---------------------------------
    const float four_s = 4.0f * scale_p[0];
    float w[K_];
    #pragma unroll
    for (int n = 0; n < K_; ++n) {
        const float z = (float)(n - HALF);
        w[n] = -(z * z) / four_s;
    }

    // ---------------- max-plus sweep: 4 output cols per thread ----------
    // output col j needs padded taps xp[j..j+14] which live at LDS j+1..j+15
    const int base = 4 * t;                // cols base .. base+3
    float acc0 = -__builtin_inff(), acc1 = acc0, acc2 = acc0, acc3 = acc0;

    #pragma unroll
    for (int c = 0; c < CIN; ++c) {
        float win[K_ + 3];                 // 18-float register window
        #pragma unroll
        for (int i = 0; i < K_ + 3; ++i)
            win[i] = s[c * SROW + base + 1 + i];
        #pragma unroll
        for (int n = 0; n < K_; ++n) {
            const float wn = w[n];
            acc0 = fmaxf(acc0, win[n]     + wn);
            acc1 = fmaxf(acc1, win[n + 1] + wn);
            acc2 = fmaxf(acc2, win[n + 2] + wn);
            acc3 = fmaxf(acc3, win[n + 3] + wn);
        }
    }

    // ---------------- broadcast to all 16 output channels (NT stores) ---
    v4f r; r.x = acc0; r.y = acc1; r.z = acc2; r.w = acc3;
    float* obase = out + ((size_t)(b * COUT) * H_ + h) * W_ + base;
    #pragma unroll
    for (int co = 0; co < COUT; ++co) {
        __builtin_nontemporal_store(r, (v4f*)(obase + (size_t)co * H_ * W_));
    }
}

extern "C" void kernel_launch(void* const* d_in, const int* in_sizes, int n_in,
                              void* d_out, int out_size, void* d_ws, size_t ws_size,
                              hipStream_t stream) {
    const float* x     = (const float*)d_in[0];
    const float* scale = (const float*)d_in[1];
    float*       out   = (float*)d_out;

    dim3 grid(B_ * H_);   // 8192 workgroups, one per (b, h) row
    dim3 block(BLOCK);
    semiconv_maxplus_kernel<<<grid, block, 0, stream>>>(x, scale, out);
}